// SSIMLoss_50775103373673
// MI455X (gfx1250) — compile-verified
//
#include <hip/hip_runtime.h>

// ---------------------------------------------------------------------------
// SSIM loss, fully fused, FP32 WMMA-based separable Gaussian blur (gfx1250)
//
//   blur = banded GEMM: Out[16x16] = Gv[16x28] x ( In[28x28] x Gh[28x16] )
//   done with V_WMMA_F32_16X16X4_F32 (16x16 tile, K in steps of 4)
// ---------------------------------------------------------------------------

typedef __attribute__((ext_vector_type(2))) float v2f;
typedef __attribute__((ext_vector_type(8))) float v8f;

#define H_IMG 512
#define W_IMG 512
#define N_IMG 64
#define TILE  16
#define KWIN  11
#define PAD   5
#define REG   26   // TILE + KWIN - 1
#define SROWS 32   // REG padded up (2 WMMA M-tiles)
#define SSTR  28   // REG padded to multiple of 4 (WMMA K)
#define HSTR  17   // padded column stride for intermediate planes

// Gaussian window, sigma = 1.5, size 11, normalized (matches reference).
__constant__ float c_g[KWIN] = {
    0.00102838f, 0.00759875f, 0.03600077f, 0.10936069f, 0.21300553f,
    0.26601172f,
    0.21300553f, 0.10936069f, 0.03600077f, 0.00759875f, 0.00102838f};

__device__ __forceinline__ float gband(int d) {
    return (d >= 0 && d < KWIN) ? c_g[d] : 0.0f;
}

__device__ __forceinline__ v8f wmma4(v2f a, v2f b, v8f c) {
    // D = A(16x4,f32) * B(4x16,f32) + C(16x16,f32)
    return __builtin_amdgcn_wmma_f32_16x16x4_f32(
        /*neg_a=*/false, a, /*neg_b=*/false, b,
        /*c_mod=*/(short)0, c, /*reuse_a=*/false, /*reuse_b=*/false);
}

__global__ __launch_bounds__(32)
void ssim_tile_kernel(const float* __restrict__ pred,
                      const float* __restrict__ target,
                      float* __restrict__ accum)
{
    __shared__ __align__(16) float sP[SROWS][SSTR];
    __shared__ __align__(16) float sT[SROWS][SSTR];
    __shared__ __align__(16) float sH[5][SROWS][HSTR];

    const int lane = threadIdx.x;     // 0..31 (wave32)
    const int lr   = lane & 15;       // lane within half
    const int hi   = lane >> 4;       // which half (K split for f32 WMMA)

    const int tpr  = W_IMG / TILE;            // 32 tiles per row
    const int tpi  = tpr * (H_IMG / TILE);    // 1024 tiles per image
    const int tile = blockIdx.x;
    const int img  = tile / tpi;
    const int tin  = tile % tpi;
    const int r0   = (tin / tpr) * TILE;
    const int c0   = (tin % tpr) * TILE;

    const float* __restrict__ Pimg = pred   + (size_t)img * (H_IMG * W_IMG);
    const float* __restrict__ Timg = target + (size_t)img * (H_IMG * W_IMG);

    // ---- cooperative load of 26x26 halo region, zero padded to 32x28 ----
    for (int e = lane; e < SROWS * SSTR; e += 32) {
        const int rr = e / SSTR, cc = e % SSTR;
        const int gr = r0 - PAD + rr;
        const int gc = c0 - PAD + cc;
        const bool ok = (rr < REG) && (cc < REG) &&
                        (gr >= 0) && (gr < H_IMG) && (gc >= 0) && (gc < W_IMG);
        const size_t idx = (size_t)gr * W_IMG + gc;
        sP[rr][cc] = ok ? Pimg[idx] : 0.0f;
        sT[rr][cc] = ok ? Timg[idx] : 0.0f;
    }
    __syncthreads();

    // ---- banded Gaussian operand (same per-lane values serve as B in
    //      stage 1 and as A in stage 2):  w[k][n] = g[k - n] ----
    v2f w[7];
    #pragma unroll
    for (int kk = 0; kk < 7; ++kk) {
        const int k = 4 * kk + 2 * hi;
        w[kk].x = gband(k - lr);
        w[kk].y = gband(k + 1 - lr);
    }

    const v8f vzero = {0.f, 0.f, 0.f, 0.f, 0.f, 0.f, 0.f, 0.f};

    // ---- stage 1: horizontal blur of 5 planes (p, t, p^2, t^2, p*t) ----
    v8f h[5][2];
    #pragma unroll
    for (int q = 0; q < 5; ++q) { h[q][0] = vzero; h[q][1] = vzero; }

    #pragma unroll
    for (int m = 0; m < 2; ++m) {
        #pragma unroll
        for (int kk = 0; kk < 7; ++kk) {
            const int k   = 4 * kk + 2 * hi;
            const int row = m * 16 + lr;
            const v2f p  = *(const v2f*)&sP[row][k];
            const v2f t  = *(const v2f*)&sT[row][k];
            const v2f pp = p * p;
            const v2f tt = t * t;
            const v2f pt = p * t;
            h[0][m] = wmma4(p,  w[kk], h[0][m]);
            h[1][m] = wmma4(t,  w[kk], h[1][m]);
            h[2][m] = wmma4(pp, w[kk], h[2][m]);
            h[3][m] = wmma4(tt, w[kk], h[3][m]);
            h[4][m] = wmma4(pt, w[kk], h[4][m]);
        }
    }

    // ---- re-layout D -> B through LDS (D: VGPR i = rows i / i+8) ----
    #pragma unroll
    for (int q = 0; q < 5; ++q)
        #pragma unroll
        for (int m = 0; m < 2; ++m)
            #pragma unroll
            for (int i = 0; i < 8; ++i)
                sH[q][m * 16 + i + 8 * hi][lr] = h[q][m][i];
    __syncthreads();

    // ---- stage 2: vertical blur ----
    v8f d[5];
    #pragma unroll
    for (int q = 0; q < 5; ++q) d[q] = vzero;

    #pragma unroll
    for (int kk = 0; kk < 7; ++kk) {
        const int k = 4 * kk + 2 * hi;
        #pragma unroll
        for (int q = 0; q < 5; ++q) {
            v2f b;
            b.x = sH[q][k][lr];
            b.y = sH[q][k + 1][lr];
            d[q] = wmma4(w[kk], b, d[q]);
        }
    }

    // ---- SSIM map + local sum over this tile ----
    const float C1v = 1e-4f, C2v = 9e-4f;
    float s = 0.0f;
    #pragma unroll
    for (int i = 0; i < 8; ++i) {
        const float m1  = d[0][i];
        const float m2  = d[1][i];
        const float m1s = m1 * m1;
        const float m2s = m2 * m2;
        const float m12 = m1 * m2;
        const float s1  = d[2][i] - m1s;
        const float s2  = d[3][i] - m2s;
        const float s12 = d[4][i] - m12;
        const float num = (2.0f * m12 + C1v) * (2.0f * s12 + C2v);
        const float den = (m1s + m2s + C1v) * (s1 + s2 + C2v);
        s += num / den;
    }

    // ---- wave32 reduction, then one atomic per tile ----
    #pragma unroll
    for (int off = 16; off > 0; off >>= 1)
        s += __shfl_xor(s, off, 32);

    if (lane == 0)
        unsafeAtomicAdd(accum, s);
}

__global__ void ssim_zero_kernel(float* __restrict__ ws) {
    if (threadIdx.x == 0 && blockIdx.x == 0) ws[0] = 0.0f;
}

__global__ void ssim_final_kernel(const float* __restrict__ ws,
                                  float* __restrict__ out) {
    const float invN = 1.0f / (float)(N_IMG * H_IMG * W_IMG);
    out[0] = 1.0f - ws[0] * invN;
}

extern "C" void kernel_launch(void* const* d_in, const int* in_sizes, int n_in,
                              void* d_out, int out_size, void* d_ws, size_t ws_size,
                              hipStream_t stream) {
    (void)in_sizes; (void)n_in; (void)out_size; (void)ws_size;
    const float* pred   = (const float*)d_in[0];
    const float* target = (const float*)d_in[1];
    float* ws  = (float*)d_ws;
    float* out = (float*)d_out;

    const int tiles = N_IMG * (H_IMG / TILE) * (W_IMG / TILE);  // 65536

    ssim_zero_kernel<<<1, 32, 0, stream>>>(ws);
    ssim_tile_kernel<<<tiles, 32, 0, stream>>>(pred, target, ws);
    ssim_final_kernel<<<1, 1, 0, stream>>>(ws, out);
}